// AdaptiveEmbeddingT2I_61718680044418
// MI455X (gfx1250) — compile-verified
//
#include <hip/hip_runtime.h>

#define B_IMG 64
#define B_CAP 64
#define R_DIM 36
#define T_DIM 50
#define D_DIM 1024
#define SMOOTH 10.0f
#define EPS_BN 1e-5f
#define EPS_L2 1e-8f

typedef float v2f __attribute__((ext_vector_type(2)));
typedef float v8f __attribute__((ext_vector_type(8)));

// ---------------------------------------------------------------------------
// Stage 1: BatchNorm1d training-mode stats per channel d over (B_IMG, R) =
// 2304 samples. 1024 threads, one per channel; loads coalesce across threads.
// ---------------------------------------------------------------------------
__global__ __launch_bounds__(256) void bn_stats_kernel(
    const float* __restrict__ img, float* __restrict__ mu, float* __restrict__ rstd) {
  int d = blockIdx.x * blockDim.x + threadIdx.x;   // 0..1023
  const float* p = img + d;
  float s = 0.f, ss = 0.f;
  #pragma unroll 4
  for (int j = 0; j < B_IMG * R_DIM; ++j) {
    float x = p[(size_t)j * D_DIM];
    s += x;
    ss = fmaf(x, x, ss);
  }
  const float inv = 1.0f / (float)(B_IMG * R_DIM);
  float m = s * inv;
  float v = ss * inv - m * m;
  mu[d] = m;
  rstd[d] = rsqrtf(v + EPS_BN);
}

// ---------------------------------------------------------------------------
// Stage 2: masked caption mean over first lens[c] words + ||cap_repr||_2.
// One block per caption, 256 threads, 4 channels per thread.
// ---------------------------------------------------------------------------
__global__ __launch_bounds__(256) void cap_repr_kernel(
    const float* __restrict__ cap, const int* __restrict__ lens,
    float* __restrict__ cap_repr, float* __restrict__ cap_norm) {
  int c = blockIdx.x;
  int len = lens[c];
  float invl = 1.0f / (float)len;
  float ssq = 0.f;
  for (int d = threadIdx.x; d < D_DIM; d += blockDim.x) {
    const float* p = cap + ((size_t)c * T_DIM) * D_DIM + d;
    float s = 0.f;
    for (int t = 0; t < len; ++t) s += p[(size_t)t * D_DIM];
    float v = s * invl;
    cap_repr[c * D_DIM + d] = v;
    ssq = fmaf(v, v, ssq);
  }
  __shared__ float red[8];
  #pragma unroll
  for (int o = 16; o > 0; o >>= 1) ssq += __shfl_down(ssq, o, 32);
  if ((threadIdx.x & 31) == 0) red[threadIdx.x >> 5] = ssq;
  __syncthreads();
  if (threadIdx.x == 0) {
    float t = 0.f;
    #pragma unroll
    for (int w = 0; w < 8; ++w) t += red[w];
    cap_norm[c] = sqrtf(t) + EPS_L2;
  }
}

// ---------------------------------------------------------------------------
// Stage 3: FiLM parameter GEMMs via V_WMMA_F32_16X16X4_F32 (full fp32).
//   out[c,n] = sum_k cap_repr[c,k] * W[n,k] + bias[n]
// One wave per 16x16 tile; K-loop in steps of 4; 256 WMMAs per tile.
// Operand layouts per CDNA5 ISA 7.12.2:
//   A (16x4, 32-bit): vgpr v holds K = k0 + v + 2*(lane>>4), row M = lane&15
//   B (4x16, 32-bit): vgpr v holds K = k0 + v + 2*(lane>>4), col N = lane&15
//   C/D (16x16):      vgpr i holds row M = i + 8*(lane>>4),  col N = lane&15
// blockIdx.z selects gamma (0) vs beta (1) weights/outputs.
// ---------------------------------------------------------------------------
__global__ __launch_bounds__(32) void film_gemm_kernel(
    const float* __restrict__ A,
    const float* __restrict__ Wg, const float* __restrict__ bg,
    const float* __restrict__ Wb, const float* __restrict__ bb,
    float* __restrict__ Gout, float* __restrict__ Bout) {
  const float* W  = blockIdx.z ? Wb : Wg;
  const float* bi = blockIdx.z ? bb : bg;
  float* out      = blockIdx.z ? Bout : Gout;

  int n0 = blockIdx.x * 16;
  int m0 = blockIdx.y * 16;
  int lane = threadIdx.x;
  int half = lane >> 4;
  int l    = lane & 15;

  const float* arow = A + (size_t)(m0 + l) * D_DIM;
  const float* wrow = W + (size_t)(n0 + l) * D_DIM;

  v8f acc = {0.f, 0.f, 0.f, 0.f, 0.f, 0.f, 0.f, 0.f};
  for (int k = 0; k < D_DIM; k += 4) {
    int ka = k + 2 * half;
    v2f av, bv;
    av.x = arow[ka];
    av.y = arow[ka + 1];
    bv.x = wrow[ka];
    bv.y = wrow[ka + 1];
    acc = __builtin_amdgcn_wmma_f32_16x16x4_f32(
        /*neg_a=*/false, av, /*neg_b=*/false, bv,
        /*c_mod=*/(short)0, acc, /*reuse_a=*/false, /*reuse_b=*/false);
  }
  float bval = bi[n0 + l];
  #pragma unroll
  for (int i = 0; i < 8; ++i) {
    out[(size_t)(m0 + i + 8 * half) * D_DIM + n0 + l] = acc[i] + bval;
  }
}

// ---------------------------------------------------------------------------
// Stage 4: FiLM modulate + fovea softmax + l2norm + cosine similarity.
// Block = (image b, 16-caption tile). The whole normalized image slice
// (36 x 1024 fp32 = 144 KB) is staged once into dynamic LDS (320 KB/WGP on
// CDNA5) and reused by all 16 captions. LDS layout [r*D + d] keeps each
// lane on a fixed bank (d % 64) across the region loop -> conflict-free.
// Two-pass numerically-stable softmax over the 36 regions per (c,d),
// recomputing mod from LDS (an fma) instead of keeping 36 live registers.
// ---------------------------------------------------------------------------
__global__ __launch_bounds__(256) void fovea_kernel(
    const float* __restrict__ img,
    const float* __restrict__ mu, const float* __restrict__ rstd,
    const float* __restrict__ cap_repr, const float* __restrict__ cap_norm,
    const float* __restrict__ gammas, const float* __restrict__ betas,
    float* __restrict__ out) {
  extern __shared__ float xs[];   // R_DIM * D_DIM floats
  int b   = blockIdx.x;
  int ct0 = blockIdx.y * 16;

  const float* ib = img + (size_t)b * R_DIM * D_DIM;
  for (int idx = threadIdx.x; idx < R_DIM * D_DIM; idx += blockDim.x) {
    int d = idx & (D_DIM - 1);
    xs[idx] = (ib[idx] - mu[d]) * rstd[d];
  }
  __syncthreads();

  __shared__ float red_dot[8];
  __shared__ float red_ss[8];

  for (int cc = 0; cc < 16; ++cc) {
    int c = ct0 + cc;
    float dot = 0.f, ss = 0.f;
    for (int d = threadIdx.x; d < D_DIM; d += blockDim.x) {
      float g  = gammas[(size_t)c * D_DIM + d] + 1.0f;
      float be = betas[(size_t)c * D_DIM + d];
      const float* xp = xs + d;
      // pass 1: max of SMOOTH*mod over regions
      float mx = -__builtin_inff();
      #pragma unroll
      for (int r = 0; r < R_DIM; ++r) {
        float mod = fmaf(xp[r * D_DIM], g, be);
        mx = fmaxf(mx, mod * SMOOTH);
      }
      // pass 2: exp-sum and weighted sum
      float se = 0.f, sem = 0.f;
      #pragma unroll
      for (int r = 0; r < R_DIM; ++r) {
        float mod = fmaf(xp[r * D_DIM], g, be);
        float e = __expf(fmaf(mod, SMOOTH, -mx));
        se += e;
        sem = fmaf(e, mod, sem);
      }
      float iv = sem / (se * (float)R_DIM);   // (w*mod).mean over regions
      dot = fmaf(iv, cap_repr[(size_t)c * D_DIM + d], dot);
      ss  = fmaf(iv, iv, ss);
    }
    #pragma unroll
    for (int o = 16; o > 0; o >>= 1) {
      dot += __shfl_down(dot, o, 32);
      ss  += __shfl_down(ss, o, 32);
    }
    if ((threadIdx.x & 31) == 0) {
      red_dot[threadIdx.x >> 5] = dot;
      red_ss[threadIdx.x >> 5]  = ss;
    }
    __syncthreads();
    if (threadIdx.x == 0) {
      float td = 0.f, ts = 0.f;
      #pragma unroll
      for (int w = 0; w < 8; ++w) { td += red_dot[w]; ts += red_ss[w]; }
      float niv = sqrtf(ts) + EPS_L2;
      out[b * B_CAP + c] = td / (niv * cap_norm[c]);
    }
    __syncthreads();
  }
}

// ---------------------------------------------------------------------------
// Launch
// ---------------------------------------------------------------------------
extern "C" void kernel_launch(void* const* d_in, const int* in_sizes, int n_in,
                              void* d_out, int out_size, void* d_ws, size_t ws_size,
                              hipStream_t stream) {
  const float* img  = (const float*)d_in[0];   // (64, 36, 1024)
  const float* cap  = (const float*)d_in[1];   // (64, 50, 1024)
  const int*   lens = (const int*)d_in[2];     // (64,)
  const float* Wg   = (const float*)d_in[3];   // (1024, 1024)
  const float* bg   = (const float*)d_in[4];   // (1024,)
  const float* Wb   = (const float*)d_in[5];   // (1024, 1024)
  const float* bb   = (const float*)d_in[6];   // (1024,)
  float* out = (float*)d_out;                  // (64, 64)

  float* ws       = (float*)d_ws;
  float* mu       = ws;                        // 1024
  float* rstd     = mu + D_DIM;                // 1024
  float* cap_rep  = rstd + D_DIM;              // 64*1024
  float* cap_nrm  = cap_rep + B_CAP * D_DIM;   // 64
  float* gam      = cap_nrm + B_CAP;           // 64*1024
  float* bet      = gam + B_CAP * D_DIM;       // 64*1024

  bn_stats_kernel<<<dim3(D_DIM / 256), 256, 0, stream>>>(img, mu, rstd);
  cap_repr_kernel<<<dim3(B_CAP), 256, 0, stream>>>(cap, lens, cap_rep, cap_nrm);
  film_gemm_kernel<<<dim3(D_DIM / 16, B_CAP / 16, 2), 32, 0, stream>>>(
      cap_rep, Wg, bg, Wb, bb, gam, bet);
  fovea_kernel<<<dim3(B_IMG, B_CAP / 16), 256,
                 (size_t)(R_DIM * D_DIM * sizeof(float)), stream>>>(
      img, mu, rstd, cap_rep, cap_nrm, gam, bet, out);
}